// DDDDepthDiff_65000035058615
// MI455X (gfx1250) — compile-verified
//
#include <hip/hip_runtime.h>
#include <stdint.h>
#include <math.h>

// NYU/Kinect intrinsics
#define KFX 582.6244816773795f
#define KFY 582.6910327098864f
#define KCX 313.0447587080473f
#define KCY 238.44389626620386f

constexpr int IMG_W = 640;
constexpr int IMG_H = 480;
constexpr int BATCH = 64;
constexpr int NELEM = BATCH * IMG_H * IMG_W;        // 19,660,800
constexpr int QUADS = NELEM / 4;                    // 4,915,200 float4s
constexpr int TPB = 256;
constexpr int WAVES_PER_BLOCK = TPB / 32;           // 8 (wave32)
constexpr int BLOCKS = 1200;
constexpr int TOTAL_WAVES = BLOCKS * WAVES_PER_BLOCK; // 9600
constexpr int TILES = QUADS / 32;                   // 153,600 (512B per array per wave-tile)
constexpr int ITERS = TILES / TOTAL_WAVES;          // 16, exact
constexpr int STAGES = 4;                           // async pipeline depth (3 ahead)

static_assert(TILES % TOTAL_WAVES == 0, "tile mapping must be exact");

__global__ __launch_bounds__(TPB) void depth_loss_stage1(
    const float* __restrict__ fake, const float* __restrict__ real,
    float* __restrict__ partials)
{
    // Per-wave private 4-stage (STAGES) ring buffers, filled by the async DMA engine.
    __shared__ float4 lds_f[STAGES][WAVES_PER_BLOCK][32];
    __shared__ float4 lds_r[STAGES][WAVES_PER_BLOCK][32];
    __shared__ float red[WAVES_PER_BLOCK][5];

    const int lane  = threadIdx.x & 31;
    const int wave  = threadIdx.x >> 5;
    const int gwave = blockIdx.x * WAVES_PER_BLOCK + wave;

    // Issue one async (fake,real) pair for a given tile stage into buffer `buf`.
    auto issue = [&](int stage, int buf) {
        const int tile = gwave + stage * TOTAL_WAVES;
        const float* gf = fake + (size_t)tile * 128 + (size_t)lane * 4;
        const float* gr = real + (size_t)tile * 128 + (size_t)lane * 4;
        // Generic shared pointer low 32 bits == LDS byte offset.
        uint32_t lf = (uint32_t)(uintptr_t)&lds_f[buf][wave][lane];
        uint32_t lr = (uint32_t)(uintptr_t)&lds_r[buf][wave][lane];
        asm volatile("global_load_async_to_lds_b128 %0, %1, off"
                     :: "v"(lf), "v"((uint64_t)(uintptr_t)gf) : "memory");
        asm volatile("global_load_async_to_lds_b128 %0, %1, off"
                     :: "v"(lr), "v"((uint64_t)(uintptr_t)gr) : "memory");
    };

    // Prologue: prime the pipeline with STAGES-1 tiles.
    #pragma unroll
    for (int s = 0; s < STAGES - 1; ++s) issue(s, s);

    float cnt = 0.f, sx = 0.f, sy = 0.f, sz = 0.f, sl = 0.f;

    for (int i = 0; i < ITERS; ++i) {
        // Make sure the previous iteration's ds_loads have drained before the
        // DMA engine may overwrite that buffer.
        asm volatile("s_wait_dscnt 0" ::: "memory");
        int ps = i + (STAGES - 1);
        if (ps > ITERS - 1) ps = ITERS - 1;           // clamped tail keeps issue count uniform
        issue(ps, (i + (STAGES - 1)) % STAGES);
        // Exactly one pair issued per iteration => allowing 6 outstanding ops
        // (3 newest pairs) guarantees pair i is complete (in-order completion).
        asm volatile("s_wait_asynccnt 6" ::: "memory");

        const float4 fv = lds_f[i % STAGES][wave][lane];
        const float4 rv = lds_r[i % STAGES][wave][lane];

        const int tile = gwave + i * TOTAL_WAVES;
        const uint32_t p0   = (uint32_t)tile * 128u + (uint32_t)lane * 4u; // flat float index
        const uint32_t col0 = p0 % (uint32_t)IMG_W;   // quad never crosses a row (640 % 4 == 0)
        const uint32_t row  = (p0 / (uint32_t)IMG_W) % (uint32_t)IMG_H;
        const float ky  = ((float)row - KCY) * (1.0f / KFY);
        const float ky2 = ky * ky;

        const float fa[4] = { fv.x, fv.y, fv.z, fv.w };
        const float ra[4] = { rv.x, rv.y, rv.z, rv.w };
        #pragma unroll
        for (int j = 0; j < 4; ++j) {
            const float kx = ((float)(col0 + (uint32_t)j) - KCX) * (1.0f / KFX);
            const float r = ra[j], f = fa[j];
            const bool m = (r > 0.f) & (r < 1.f) & (f > 0.f) & (f < 1.f);
            const float d  = r - f;
            const float d2 = d * d;
            const float dl = __log2f(r) - __log2f(f); // NaN outside mask -> select below
            cnt += m ? 1.0f       : 0.0f;
            sz  += m ? d2         : 0.0f;
            sx  += m ? d2 * kx*kx : 0.0f;
            sy  += m ? d2 * ky2   : 0.0f;
            sl  += m ? dl * dl    : 0.0f;
        }
    }
    // log2 -> ln conversion applied once: (ln2)^2
    sl *= 0.48045301391820142f;

    // Wave32 butterfly reduction (deterministic).
    float v[5] = { cnt, sx, sy, sz, sl };
    #pragma unroll
    for (int c = 0; c < 5; ++c)
        #pragma unroll
        for (int off = 16; off > 0; off >>= 1)
            v[c] += __shfl_xor(v[c], off, 32);

    if (lane == 0) {
        #pragma unroll
        for (int c = 0; c < 5; ++c) red[wave][c] = v[c];
    }
    __syncthreads();
    if (threadIdx.x < 5) {
        float a = 0.f;
        #pragma unroll
        for (int w = 0; w < WAVES_PER_BLOCK; ++w) a += red[w][threadIdx.x];
        partials[(size_t)blockIdx.x * 5 + threadIdx.x] = a;
    }
}

__global__ __launch_bounds__(TPB) void depth_loss_stage2(
    const float* __restrict__ partials, float* __restrict__ out)
{
    __shared__ float sh[5][TPB];
    float a[5] = { 0.f, 0.f, 0.f, 0.f, 0.f };
    for (int i = threadIdx.x; i < BLOCKS; i += TPB) {
        #pragma unroll
        for (int c = 0; c < 5; ++c) a[c] += partials[(size_t)i * 5 + c];
    }
    #pragma unroll
    for (int c = 0; c < 5; ++c) sh[c][threadIdx.x] = a[c];
    __syncthreads();
    for (int off = TPB / 2; off > 0; off >>= 1) {
        if ((int)threadIdx.x < off) {
            #pragma unroll
            for (int c = 0; c < 5; ++c)
                sh[c][threadIdx.x] += sh[c][threadIdx.x + off];
        }
        __syncthreads();
    }
    if (threadIdx.x == 0) {
        const float n  = sh[0][0];
        const float lX = sqrtf(sh[1][0] / n);
        const float lY = sqrtf(sh[2][0] / n);
        const float lZ = sqrtf(sh[3][0] / n);
        const float rl = sqrtf(sh[4][0] / n);
        out[0] = 10.0f * (rl + fabsf(10.0f * (3.0f - expf(lX) - expf(lY) - expf(lZ))));
    }
}

extern "C" void kernel_launch(void* const* d_in, const int* in_sizes, int n_in,
                              void* d_out, int out_size, void* d_ws, size_t ws_size,
                              hipStream_t stream) {
    (void)in_sizes; (void)n_in; (void)out_size; (void)ws_size;
    const float* fake = (const float*)d_in[0];
    const float* real = (const float*)d_in[1];
    float* partials = (float*)d_ws;       // BLOCKS * 5 floats = 24 KB
    float* out = (float*)d_out;

    depth_loss_stage1<<<BLOCKS, TPB, 0, stream>>>(fake, real, partials);
    depth_loss_stage2<<<1, TPB, 0, stream>>>(partials, out);
}